// ElboLossZ_64518998721071
// MI455X (gfx1250) — compile-verified
//
#include <hip/hip_runtime.h>

typedef __attribute__((ext_vector_type(2))) float v2f;
typedef __attribute__((ext_vector_type(8))) float v8f;

// Fast hardware approximations (v_rcp_f32 / v_rsq_f32, ~1ulp).  Branchless:
// keeps the per-step masking as v_cndmask instead of exec-mask branch regions.
__device__ __forceinline__ float fast_rcp(float x) { return __builtin_amdgcn_rcpf(x); }
__device__ __forceinline__ float fast_rsq(float x) { return __builtin_amdgcn_rsqf(x); }

// ---------------------------------------------------------------------------
// Load row r (16 contiguous floats, 16B-aligned) into registers.
// ---------------------------------------------------------------------------
__device__ __forceinline__ void load_rows16(float a[16], const float* __restrict__ p) {
  const float4* q = reinterpret_cast<const float4*>(p);
#pragma unroll
  for (int i = 0; i < 4; ++i) {
    float4 t = q[i];
    a[4 * i + 0] = t.x;
    a[4 * i + 1] = t.y;
    a[4 * i + 2] = t.z;
    a[4 * i + 3] = t.w;
  }
}

// ---------------------------------------------------------------------------
// Row-distributed Cholesky of a symmetric 16x16 matrix.
// Lane r (within its 16-lane half) holds row r in a[0..15].
// Returns log(det) = sum_k log(pivot_k).  Pivots >= 1 (A >= I), so two
// partial products stay well inside f32 range -> only 2 log calls.
// If KEEPL, on exit a[k] = L[r][k] for k <= r (lower triangle, row-local).
// ---------------------------------------------------------------------------
template <bool KEEPL>
__device__ __forceinline__ float chol16(float a[16], int r) {
  float p0 = 1.0f, p1 = 1.0f;
#pragma unroll
  for (int k = 0; k < 16; ++k) {
    float akk = __shfl(a[k], k, 16);      // current pivot A[k][k]
    if (k < 8) p0 *= akk; else p1 *= akk;
    float ra = fast_rcp(akk);
    float f  = (r > k) ? a[k] * ra : 0.0f;   // A[r][k]/akk  (masked, branchless)
    // Trailing update: A[r][c] -= A[r][k] * A[k][c] / akk.  Lane k never
    // updates (f==0 there), so its broadcast values stay pristine.
#pragma unroll
    for (int c = k + 1; c < 16; ++c) {
      float rkc = __shfl(a[c], k, 16);    // A[k][c] == A[c][k] (symmetry)
      a[c] -= rkc * f;
    }
    if (KEEPL) {
      float rs     = fast_rsq(akk);
      float lkk    = akk * rs;            // sqrt(akk)
      float scaled = a[k] * rs;           // L[r][k] for r > k
      a[k] = (r == k) ? lkk : ((r > k) ? scaled : a[k]);
    }
  }
  return __logf(p0) + __logf(p1);
}

// ---------------------------------------------------------------------------
// Forward substitution  L y = v ;  returns ||y||^2 (uniform in the half).
// Lane r holds row r of L (a[]) and v_r.
// ---------------------------------------------------------------------------
__device__ __forceinline__ float solve16(const float a[16], float v, int r) {
  float acc = v, q = 0.0f;
#pragma unroll
  for (int k = 0; k < 16; ++k) {
    float dk = __shfl(a[k], k, 16);   // L[k][k]
    float nk = __shfl(acc, k, 16);    // residual at row k (final by step k)
    float yk = nk * fast_rcp(dk);
    q += yk * yk;
    acc -= ((r > k) ? a[k] : 0.0f) * yk;  // a[k] = L[r][k], valid for r>k
  }
  return q;
}

// ---------------------------------------------------------------------------
// g = e (1x16) @ S (16x16) via chained V_WMMA_F32_16X16X4_F32.
// A (16x4): every row = e[4j .. 4j+3]  -> all 16 rows of D are identical.
// ISA 7.12.2 32-bit A layout: VGPR0 holds K=0 (lanes 0-15) / K=2 (16-31),
// VGPR1 holds K=1 / K=3; B (4x16) uses the matching K slots, N = lane%16.
// After 4 chained WMMAs, acc[0] in EVERY lane == g[lane%16].
// EXEC is all ones here (straight-line code, full blocks).
// ---------------------------------------------------------------------------
__device__ __forceinline__ float gemv16_wmma(const float* __restrict__ S,
                                             const float* __restrict__ e,
                                             int lane, int r) {
  const int kk = (lane < 16) ? 0 : 2;
  v8f acc = {0.f, 0.f, 0.f, 0.f, 0.f, 0.f, 0.f, 0.f};
#pragma unroll
  for (int j = 0; j < 4; ++j) {
    v2f av, bv;
    av.x = e[4 * j + kk];
    av.y = e[4 * j + kk + 1];
    bv.x = S[(4 * j + kk) * 16 + r];
    bv.y = S[(4 * j + kk + 1) * 16 + r];
    acc = __builtin_amdgcn_wmma_f32_16x16x4_f32(
        /*neg_a=*/false, av, /*neg_b=*/false, bv,
        /*c_mod=*/(short)0, acc, /*reuse_a=*/false, /*reuse_b=*/false);
  }
  return acc[0];
}

// ---------------------------------------------------------------------------
// Main kernel: one wave handles 2 items (lanes 0-15 item A, 16-31 item B).
// Block = 256 threads = 8 waves = 16 items.
// ---------------------------------------------------------------------------
__global__ __launch_bounds__(256) void elbo_kernel(
    const float* __restrict__ x, const float* __restrict__ mutilda_z,
    const float* __restrict__ St, const float* __restrict__ mu_x,
    const float* __restrict__ Sx, const float* __restrict__ mu_z,
    const float* __restrict__ Sz, const float* __restrict__ eps,
    double* __restrict__ partial) {
  const int tid  = threadIdx.x;
  const int lane = tid & 31;
  const int wave = tid >> 5;
  const int r    = lane & 15;
  const int half = lane >> 4;

  const long pairBase = ((long)blockIdx.x * 8 + wave) * 2;
  const long n  = pairBase + half;   // this half's item
  const long nA = pairBase;
  const long nB = pairBase + 1;

  // ---- GEMV for both items via WMMA (straight-line, EXEC all ones) ------
  float gA = gemv16_wmma(St + nA * 256, eps + nA * 16, lane, r);
  float gB = gemv16_wmma(St + nB * 256, eps + nB * 16, lane, r);
  float g  = half ? gB : gA;         // (eps @ St)[r] for my item

  float a[16];

  // ---- part5: log det sigmatilda_z (factor only, discard L) -------------
  load_rows16(a, St + n * 256 + (long)r * 16);
  float part5 = chol16<false>(a, r);

  // ---- part1 + part2: sigma_z ------------------------------------------
  load_rows16(a, Sz + n * 256 + (long)r * 16);
  float part1 = chol16<true>(a, r);
  float d     = mutilda_z[n * 16 + r] + g - mu_z[n * 16 + r];
  float part2 = solve16(a, d, r);

  // ---- part3 + part4: sigma_x ------------------------------------------
  load_rows16(a, Sx + n * 256 + (long)r * 16);
  float part3 = chol16<true>(a, r);
  float v     = x[n * 16 + r] - mu_x[n * 16 + r];
  float part4 = solve16(a, v, r);

  float tot = part1 + part2 + part3 + part4 - part5;

  // ---- reduction: one representative lane per half, f64 accumulate -----
  double contrib = (r == 0) ? (double)tot : 0.0;
#pragma unroll
  for (int off = 16; off >= 1; off >>= 1)
    contrib += __shfl_down(contrib, off, 32);

  __shared__ double red[8];
  if (lane == 0) red[wave] = contrib;
  __syncthreads();
  if (tid == 0) {
    double s = 0.0;
#pragma unroll
    for (int w = 0; w < 8; ++w) s += red[w];
    atomicAdd(partial, s);
  }
}

__global__ void finalize_kernel(const double* __restrict__ partial,
                                float* __restrict__ out, int n) {
  out[0] = (float)(partial[0] / (2.0 * (double)n));
}

// ---------------------------------------------------------------------------
extern "C" void kernel_launch(void* const* d_in, const int* in_sizes, int n_in,
                              void* d_out, int out_size, void* d_ws, size_t ws_size,
                              hipStream_t stream) {
  const float* x         = (const float*)d_in[0];
  const float* mutilda_z = (const float*)d_in[1];
  const float* St        = (const float*)d_in[2];
  const float* mu_x      = (const float*)d_in[3];
  const float* Sx        = (const float*)d_in[4];
  const float* mu_z      = (const float*)d_in[5];
  const float* Sz        = (const float*)d_in[6];
  const float* eps       = (const float*)d_in[7];

  const int N = in_sizes[0] / 16;          // 65536
  double* partial = (double*)d_ws;

  hipMemsetAsync(d_ws, 0, sizeof(double), stream);

  const int itemsPerBlock = 16;            // 8 waves x 2 items
  const int blocks = N / itemsPerBlock;    // N divisible by 16
  elbo_kernel<<<blocks, 256, 0, stream>>>(x, mutilda_z, St, mu_x, Sx, mu_z, Sz,
                                          eps, partial);
  finalize_kernel<<<1, 1, 0, stream>>>(partial, (float*)d_out, N);
}